// model_audio_50345606644016
// MI455X (gfx1250) — compile-verified
//
#include <hip/hip_runtime.h>
#include <math.h>

typedef __attribute__((ext_vector_type(2))) float v2f;
typedef __attribute__((ext_vector_type(8))) float v8f;

#define NBATCH 8
#define LEN    160
#define CH     32
#define NTOK   5120     // LEN*CH
#define FLATK  10240    // 2*NTOK
#define FC1N   1024
#define KSPLIT 32
#define KCHUNK (FLATK / KSPLIT)   // 320 (multiple of 32 for the unrolled loop)

__device__ __forceinline__ float elu1(float y) {
    return (y > 0.0f) ? y : (__expf(y) - 1.0f);
}

// ---------------------------------------------------------------------------
// Kernel 1: conv1d(5,pad2)+BN+ELU -> conv1d(5,pad2)+BN+ELU -> f[b][l*32+c]
// One block per batch element. Everything staged in LDS.
// ---------------------------------------------------------------------------
__global__ __launch_bounds__(256)
void k_features(const float* __restrict__ x,
                const float* __restrict__ w1, const float* __restrict__ b1,
                const float* __restrict__ g1, const float* __restrict__ be1,
                const float* __restrict__ m1, const float* __restrict__ v1,
                const float* __restrict__ w2, const float* __restrict__ b2,
                const float* __restrict__ g2, const float* __restrict__ be2,
                const float* __restrict__ m2, const float* __restrict__ v2,
                float* __restrict__ f) {
    __shared__ float sx[LEN];
    __shared__ float sh1[CH][LEN];
    __shared__ float sw2[CH * CH * 5];
    const int b = blockIdx.x;
    const int tid = threadIdx.x;

    for (int i = tid; i < LEN; i += 256) sx[i] = x[b * LEN + i];
    for (int i = tid; i < CH * CH * 5; i += 256) sw2[i] = w2[i];
    __syncthreads();

    for (int i = tid; i < CH * LEN; i += 256) {
        const int c = i / LEN, l = i % LEN;
        float acc = b1[c];
#pragma unroll
        for (int t = 0; t < 5; ++t) {
            const int p = l - 2 + t;
            const float xv = (p >= 0 && p < LEN) ? sx[p] : 0.0f;
            acc = fmaf(w1[c * 5 + t], xv, acc);
        }
        const float sc = g1[c] * rsqrtf(v1[c] + 1e-5f);
        sh1[c][l] = elu1((acc - m1[c]) * sc + be1[c]);
    }
    __syncthreads();

    for (int i = tid; i < CH * LEN; i += 256) {
        const int c = i / LEN, l = i % LEN;
        float acc = b2[c];
        for (int ci = 0; ci < CH; ++ci) {
            const float* wr = &sw2[(c * CH + ci) * 5];
#pragma unroll
            for (int t = 0; t < 5; ++t) {
                const int p = l - 2 + t;
                const float hv = (p >= 0 && p < LEN) ? sh1[ci][p] : 0.0f;
                acc = fmaf(wr[t], hv, acc);
            }
        }
        const float sc = g2[c] * rsqrtf(v2[c] + 1e-5f);
        f[b * NTOK + l * CH + c] = elu1((acc - m2[c]) * sc + be2[c]);
    }
}

// ---------------------------------------------------------------------------
// Kernel 2: rank-collapsed attention.
// attn[n,:] = softmax_m(s_n * f_m), s_n = A2*f_n + C2;
// g_n = sum_m attn[n,m]*f_m;  flat[b][c*5120+n] = gamma*(av_c*g_n+bv_c)+f_n.
// Block = (batch, 256-row chunk), 512 threads: 2 threads split each row's
// m-loop (2560 elements each, float4 LDS reads -> ds_load_b128).
// ---------------------------------------------------------------------------
__global__ __launch_bounds__(512)
void k_attention(const float* __restrict__ f,
                 const float* __restrict__ wq, const float* __restrict__ bq,
                 const float* __restrict__ wk, const float* __restrict__ bk,
                 const float* __restrict__ wv, const float* __restrict__ bv,
                 const float* __restrict__ gamma,
                 float* __restrict__ flat) {
    __shared__ float sf[NTOK];
    __shared__ float red[512];
    __shared__ float psumE[512];
    __shared__ float psumW[512];
    const int b = blockIdx.x / 20;
    const int chunk = blockIdx.x % 20;
    const int tid = threadIdx.x;

    for (int i = tid; i < NTOK; i += 512) sf[i] = f[b * NTOK + i];
    __syncthreads();

    // block-wide max/min of f[b]
    float mx = -1e30f, mn = 1e30f;
    for (int i = tid; i < NTOK; i += 512) {
        const float v = sf[i];
        mx = fmaxf(mx, v);
        mn = fminf(mn, v);
    }
    red[tid] = mx; __syncthreads();
    for (int s = 256; s > 0; s >>= 1) {
        if (tid < s) red[tid] = fmaxf(red[tid], red[tid + s]);
        __syncthreads();
    }
    const float fmax_all = red[0]; __syncthreads();
    red[tid] = mn; __syncthreads();
    for (int s = 256; s > 0; s >>= 1) {
        if (tid < s) red[tid] = fminf(red[tid], red[tid + s]);
        __syncthreads();
    }
    const float fmin_all = red[0];

    // scalar collapse of the 2-channel 1x1 projections
    const float aq0 = wq[0] + wq[1], aq1 = wq[2] + wq[3];
    const float ak0 = wk[0] + wk[1], ak1 = wk[2] + wk[3];
    const float av0 = wv[0] + wv[1], av1 = wv[2] + wv[3];
    const float A2 = aq0 * ak0 + aq1 * ak1;
    const float C2 = bq[0] * ak0 + bq[1] * ak1;
    const float gm = gamma[0];

    const int tid256 = tid & 255;
    const int half = tid >> 8;                 // which m-half this thread sums
    const int n = chunk * 256 + tid256;
    const float fn = sf[n];
    const float s = A2 * fn + C2;
    const float off = (s >= 0.0f) ? s * fmax_all : s * fmin_all;

    float sum = 0.0f, wsum = 0.0f;
    const int mbeg = half * (NTOK / 2);
#pragma unroll 2
    for (int m = mbeg; m < mbeg + NTOK / 2; m += 4) {
        const float4 fv = *reinterpret_cast<const float4*>(&sf[m]);
        const float e0 = __expf(fmaf(s, fv.x, -off));
        const float e1 = __expf(fmaf(s, fv.y, -off));
        const float e2 = __expf(fmaf(s, fv.z, -off));
        const float e3 = __expf(fmaf(s, fv.w, -off));
        sum += (e0 + e1) + (e2 + e3);
        wsum = fmaf(fv.x, e0, wsum);
        wsum = fmaf(fv.y, e1, wsum);
        wsum = fmaf(fv.z, e2, wsum);
        wsum = fmaf(fv.w, e3, wsum);
    }
    psumE[tid] = sum;
    psumW[tid] = wsum;
    __syncthreads();

    if (tid < 256) {
        const float S = psumE[tid] + psumE[tid + 256];
        const float W = psumW[tid] + psumW[tid + 256];
        const float g = W / S;
        const int nn = chunk * 256 + tid;
        const float fnn = sf[nn];
        flat[b * FLATK + nn]        = fmaf(gm, fmaf(av0, g, bv[0]), fnn);
        flat[b * FLATK + NTOK + nn] = fmaf(gm, fmaf(av1, g, bv[1]), fnn);
    }
}

// ---------------------------------------------------------------------------
// Kernel 2b: zero the M=8..15 pad rows of flat16 so the WMMA A-tile is exact
// and EXEC stays all-1s inside the WMMA loop (ISA requirement).
// ---------------------------------------------------------------------------
__global__ __launch_bounds__(256)
void k_zero_pad(float* __restrict__ flat) {
    const int i = blockIdx.x * 256 + threadIdx.x;
    if (i < NBATCH * FLATK) flat[NBATCH * FLATK + i] = 0.0f;
}

// ---------------------------------------------------------------------------
// Kernel 3: FC1 partials via V_WMMA_F32_16X16X4_F32, K-split for MLP.
// grid = (8 col-tile groups, KSPLIT k-chunks), 8 waves/block, each wave owns
// one 16-wide column tile over a 320-element K chunk (80 WMMA steps).
// Partial C tiles land in zpart[kc][16][1024]; no atomics (deterministic).
// A 16x4 layout: lanes 0-15 M=0..15 K={0,1}; lanes 16-31 K={2,3}.
// ---------------------------------------------------------------------------
__global__ __launch_bounds__(256)
void k_fc1_wmma(const float* __restrict__ flat16,   // 16 x 10240 (rows 8..15 zero)
                const float* __restrict__ fw1,      // 1024 x 10240
                float* __restrict__ zpart) {        // KSPLIT x 16 x 1024
    const int tid  = threadIdx.x;
    const int wave = tid >> 5;
    const int lane = tid & 31;
    const int l16  = lane & 15;
    const int hi   = lane >> 4;
    const int o    = (blockIdx.x * 8 + wave) * 16 + l16;
    const int kc   = blockIdx.y;
    const int kbase = kc * KCHUNK;

    const float* __restrict__ arow = flat16 + (size_t)l16 * FLATK + kbase + 2 * hi;
    const float* __restrict__ brow = fw1    + (size_t)o   * FLATK + kbase + 2 * hi;

    v8f acc = {};
#pragma unroll 8
    for (int k = 0; k < KCHUNK; k += 4) {
        const float2 av = *reinterpret_cast<const float2*>(arow + k);
        const float2 bv = *reinterpret_cast<const float2*>(brow + k);
        v2f a;  a.x  = av.x; a.y  = av.y;
        v2f bb; bb.x = bv.x; bb.y = bv.y;
        acc = __builtin_amdgcn_wmma_f32_16x16x4_f32(
            /*neg_a=*/false, a, /*neg_b=*/false, bb,
            /*c_mod=*/(short)0, acc, /*reuse_a=*/false, /*reuse_b=*/false);
    }

    float* __restrict__ zp = zpart + (size_t)kc * (16 * FC1N);
#pragma unroll
    for (int r = 0; r < 8; ++r) {
        const int m = r + 8 * hi;            // C/D: VGPR r = rows r / r+8
        zp[m * FC1N + o] = acc[r];
    }
}

// ---------------------------------------------------------------------------
// Kernel 3b: reduce the KSPLIT partials, add bias, BN, ELU -> x1 (8x1024).
// ---------------------------------------------------------------------------
__global__ __launch_bounds__(256)
void k_fc1_finish(const float* __restrict__ zpart,
                  const float* __restrict__ fb1,
                  const float* __restrict__ g6, const float* __restrict__ be6,
                  const float* __restrict__ m6, const float* __restrict__ v6,
                  float* __restrict__ x1out) {     // 8 x 1024
    const int idx = blockIdx.x * 256 + threadIdx.x;   // 0..8191
    const int o = idx & (FC1N - 1);
    const int m = idx >> 10;
    float z = fb1[o];
#pragma unroll 4
    for (int kc = 0; kc < KSPLIT; ++kc)
        z += zpart[(size_t)kc * (16 * FC1N) + m * FC1N + o];
    const float sc = g6[o] * rsqrtf(v6[o] + 1e-5f);
    x1out[idx] = elu1((z - m6[o]) * sc + be6[o]);
}

// ---------------------------------------------------------------------------
// Kernel 4: FC2 (1024 -> 13) + row softmax. One block.
// ---------------------------------------------------------------------------
__global__ __launch_bounds__(128)
void k_fc2_softmax(const float* __restrict__ x1,     // 8 x 1024
                   const float* __restrict__ fw2,    // 13 x 1024
                   const float* __restrict__ fb2,
                   float* __restrict__ probs) {      // 8 x 13
    __shared__ float z[NBATCH][13];
    const int tid = threadIdx.x;
    if (tid < NBATCH * 13) {
        const int b = tid / 13, j = tid % 13;
        float acc = fb2[j];
        const float* xr = x1 + b * FC1N;
        const float* wr = fw2 + j * FC1N;
#pragma unroll 4
        for (int k = 0; k < FC1N; ++k) acc = fmaf(xr[k], wr[k], acc);
        z[b][j] = acc;
    }
    __syncthreads();
    if (tid < NBATCH) {
        const int b = tid;
        float mx = z[b][0];
        for (int j = 1; j < 13; ++j) mx = fmaxf(mx, z[b][j]);
        float e[13], sum = 0.0f;
        for (int j = 0; j < 13; ++j) { e[j] = __expf(z[b][j] - mx); sum += e[j]; }
        const float inv = 1.0f / sum;
        for (int j = 0; j < 13; ++j) probs[b * 13 + j] = e[j] * inv;
    }
}

// ---------------------------------------------------------------------------
extern "C" void kernel_launch(void* const* d_in, const int* in_sizes, int n_in,
                              void* d_out, int out_size, void* d_ws, size_t ws_size,
                              hipStream_t stream) {
    const float* x   = (const float*)d_in[0];
    const float* w1  = (const float*)d_in[1];
    const float* b1  = (const float*)d_in[2];
    const float* g1  = (const float*)d_in[3];
    const float* be1 = (const float*)d_in[4];
    const float* m1  = (const float*)d_in[5];
    const float* v1  = (const float*)d_in[6];
    const float* w2  = (const float*)d_in[7];
    const float* b2  = (const float*)d_in[8];
    const float* g2  = (const float*)d_in[9];
    const float* be2 = (const float*)d_in[10];
    const float* m2  = (const float*)d_in[11];
    const float* v2  = (const float*)d_in[12];
    const float* wq  = (const float*)d_in[13];
    const float* bq  = (const float*)d_in[14];
    const float* wk  = (const float*)d_in[15];
    const float* bk  = (const float*)d_in[16];
    const float* wv  = (const float*)d_in[17];
    const float* bv  = (const float*)d_in[18];
    const float* gam = (const float*)d_in[19];
    const float* fw1 = (const float*)d_in[20];
    const float* fb1 = (const float*)d_in[21];
    const float* g6  = (const float*)d_in[22];
    const float* be6 = (const float*)d_in[23];
    const float* m6  = (const float*)d_in[24];
    const float* v6  = (const float*)d_in[25];
    const float* fw2 = (const float*)d_in[26];
    const float* fb2 = (const float*)d_in[27];

    float* f      = (float*)d_ws;                 // 8 x 5120
    float* flat16 = f + NBATCH * NTOK;            // 16 x 10240 (rows 8..15 zero)
    float* zpart  = flat16 + 16 * FLATK;          // KSPLIT x 16 x 1024

    float* probs = (float*)d_out;                 // 8 x 13   (output 0)
    float* x1    = probs + NBATCH * 13;           // 8 x 1024 (output 1)

    k_features<<<dim3(NBATCH), dim3(256), 0, stream>>>(
        x, w1, b1, g1, be1, m1, v1, w2, b2, g2, be2, m2, v2, f);

    k_zero_pad<<<dim3((NBATCH * FLATK + 255) / 256), dim3(256), 0, stream>>>(flat16);

    k_attention<<<dim3(NBATCH * 20), dim3(512), 0, stream>>>(
        f, wq, bq, wk, bk, wv, bv, gam, flat16);

    k_fc1_wmma<<<dim3(8, KSPLIT), dim3(256), 0, stream>>>(flat16, fw1, zpart);

    k_fc1_finish<<<dim3((NBATCH * FC1N) / 256), dim3(256), 0, stream>>>(
        zpart, fb1, g6, be6, m6, v6, x1);

    k_fc2_softmax<<<dim3(1), dim3(128), 0, stream>>>(x1, fw2, fb2, probs);
}